// BiLSTM_34711925686917
// MI455X (gfx1250) — compile-verified
//
#include <hip/hip_runtime.h>
#include <hip/hip_bf16.h>

#define HID   5
#define GATES 20      // 4*H
#define DDIM  1000
#define BATCH 1200
#define SEQ   100
#define NTILE 75      // BATCH/16

typedef float v2f __attribute__((ext_vector_type(2)));
typedef float v8f __attribute__((ext_vector_type(8)));

__device__ __forceinline__ float sigm(float x) {
    return 1.0f / (1.0f + __expf(-x));
}
__device__ __forceinline__ float tanh_fast(float x) {
    float e = __expf(2.0f * x);
    return (e - 1.0f) / (e + 1.0f);
}

// ---------------------------------------------------------------------------
// Kernel 1: gate projection GEMM  out[s][b][g] = X[b][s][:] . W[g][:] + b1[g]+b2[g]
// One wave computes a 16(b) x 20(g) tile with V_WMMA_F32_16X16X4_F32,
// looping K=DDIM in chunks of 4. W is staged in LDS as K-pair-interleaved
// transpose: wt2[p][n] = {W[n][2p], W[n][2p+1]}  (conflict-free B reads).
// The second (hi) 16-wide N tile only has 4 live columns (16..19); dead lanes
// read a duplicate of a live column instead of being predicated, so EXEC
// stays all-ones and the inner loop has no exec save/restore or zero fills.
// ---------------------------------------------------------------------------
__global__ __launch_bounds__(256) void gate_gemm(
    const float* __restrict__ X,    // [BATCH][SEQ][DDIM]
    const float* __restrict__ W,    // [GATES][DDIM]
    const float* __restrict__ b1,   // [GATES]
    const float* __restrict__ b2,   // [GATES]
    float* __restrict__ out,        // [gridDim.y][BATCH][GATES]
    int sFixed)                     // -1: s = blockIdx.y ; else fixed timestep
{
    __shared__ float wt2[(DDIM / 2) * GATES * 2];   // 80 KB

    // cooperative staging of W into pair-interleaved transposed LDS
    const int total = (DDIM / 2) * GATES * 2;
    for (int t = threadIdx.x; t < total; t += 256) {
        int half = t & 1;
        int n    = (t >> 1) % GATES;
        int p    = t / (2 * GATES);
        wt2[t]   = W[n * DDIM + 2 * p + half];
    }
    __syncthreads();

    const int lane = threadIdx.x & 31;
    const int wid  = threadIdx.x >> 5;
    const int tile = blockIdx.x * 8 + wid;
    if (tile >= NTILE) return;

    const int s  = (sFixed < 0) ? (int)blockIdx.y : sFixed;
    const int b0 = tile * 16;
    const int n  = lane & 15;              // output column (gate) 0..15
    const int hiLane = lane >> 4;          // 0: K0/K1 half, 1: K2/K3 half
    const int n2 = 16 + n;                 // true hi column
    // hi-tile column actually read: valid for n<4, duplicate (discarded) else
    const int nhi = (n2 < GATES) ? n2 : n;

    // this lane's A row: X[b0+n][s][:], pre-offset by K sub-lane (0 or 2)
    const float* xrow = X + ((size_t)(b0 + n) * SEQ + s) * DDIM + hiLane * 2;

    // LDS float-index bases; per-chunk index = k0*GATES + base
    // (addr(p,col) = (p*GATES+col)*2 floats, p = k0/2 + hiLane)
    const int baseLo = (hiLane * GATES + n)   * 2;
    const int baseHi = (hiLane * GATES + nhi) * 2;

    v8f acc_lo = {};
    v8f acc_hi = {};

    // K loop: 25 blocks of 40 (10 WMMA chunks each)
    for (int kb = 0; kb < DDIM; kb += 40) {
        __builtin_prefetch(xrow + kb + 160, 0, 3);   // stream-ahead hint
#pragma unroll
        for (int c = 0; c < 10; ++c) {
            const int k0 = kb + 4 * c;
            // A fragment: lanes 0-15 -> {K0,K1}, lanes 16-31 -> {K2,K3}
            v2f a  = *(const v2f*)(xrow + k0);
            // B fragments (unconditional, conflict-free)
            v2f bl = *(const v2f*)&wt2[k0 * GATES + baseLo];
            v2f bh = *(const v2f*)&wt2[k0 * GATES + baseHi];
            acc_lo = __builtin_amdgcn_wmma_f32_16x16x4_f32(
                false, a, false, bl, (short)0, acc_lo, false, false);
            acc_hi = __builtin_amdgcn_wmma_f32_16x16x4_f32(
                false, a, false, bh, (short)0, acc_hi, false, false);
        }
    }

    // epilogue: add fused biases, scatter 16x20 tile (hi stores only n2<20)
    float* oslab = out + (size_t)blockIdx.y * BATCH * GATES + (size_t)b0 * GATES;
    const float bias_lo = b1[n] + b2[n];
    const float bias_hi = (n2 < GATES) ? (b1[n2] + b2[n2]) : 0.0f;
    const int mofs = hiLane * 8;
#pragma unroll
    for (int r = 0; r < 8; ++r) {
        const int m = r + mofs;
        oslab[m * GATES + n] = acc_lo[r] + bias_lo;
        if (n2 < GATES)
            oslab[m * GATES + n2] = acc_hi[r] + bias_hi;
    }
}

// ---------------------------------------------------------------------------
// Kernel 2: forward LSTM scan. Batch rows are independent -> one thread per b,
// W_hh (20x5) register-resident, 100 sequential cell steps.
// ---------------------------------------------------------------------------
__global__ __launch_bounds__(256) void lstm_scan_fwd(
    const float* __restrict__ xp,   // [SEQ][BATCH][GATES]
    const float* __restrict__ h0,   // [2][BATCH][HID] (dir 0)
    const float* __restrict__ c0,
    const float* __restrict__ w_hh, // [GATES][HID]
    float* __restrict__ hcat)       // [BATCH][2*HID]
{
    const int b = blockIdx.x * blockDim.x + threadIdx.x;
    if (b >= BATCH) return;

    float w[GATES][HID];
#pragma unroll
    for (int g = 0; g < GATES; ++g)
#pragma unroll
        for (int j = 0; j < HID; ++j) w[g][j] = w_hh[g * HID + j];

    float h[HID], c[HID];
#pragma unroll
    for (int j = 0; j < HID; ++j) {
        h[j] = h0[b * HID + j];
        c[j] = c0[b * HID + j];
    }

    for (int s = 0; s < SEQ; ++s) {
        const float* xr = xp + ((size_t)s * BATCH + b) * GATES;
        float g[GATES];
#pragma unroll
        for (int gi = 0; gi < GATES; ++gi) {
            float acc = xr[gi];
#pragma unroll
            for (int j = 0; j < HID; ++j) acc = fmaf(h[j], w[gi][j], acc);
            g[gi] = acc;
        }
#pragma unroll
        for (int j = 0; j < HID; ++j) {
            float ig = sigm(g[j]);
            float fg = sigm(g[HID + j]);
            float gg = tanh_fast(g[2 * HID + j]);
            float og = sigm(g[3 * HID + j]);
            c[j] = fg * c[j] + ig * gg;
            h[j] = og * tanh_fast(c[j]);
        }
    }
#pragma unroll
    for (int j = 0; j < HID; ++j) hcat[b * (2 * HID) + j] = h[j];
}

// ---------------------------------------------------------------------------
// Kernel 3: backward direction = single cell step at t = S-1.
// ---------------------------------------------------------------------------
__global__ __launch_bounds__(256) void bwd_cell(
    const float* __restrict__ gatesb, // [BATCH][GATES] (biases already fused)
    const float* __restrict__ h0,     // [2][BATCH][HID]
    const float* __restrict__ c0,
    const float* __restrict__ w_hh,   // [GATES][HID]
    float* __restrict__ hcat)         // [BATCH][2*HID]
{
    const int b = blockIdx.x * blockDim.x + threadIdx.x;
    if (b >= BATCH) return;

    const float* h1 = h0 + (size_t)BATCH * HID;   // direction 1
    const float* c1 = c0 + (size_t)BATCH * HID;
    const float* xr = gatesb + (size_t)b * GATES;

    float h[HID], c[HID];
#pragma unroll
    for (int j = 0; j < HID; ++j) { h[j] = h1[b * HID + j]; c[j] = c1[b * HID + j]; }

    float g[GATES];
#pragma unroll
    for (int gi = 0; gi < GATES; ++gi) {
        float acc = xr[gi];
#pragma unroll
        for (int j = 0; j < HID; ++j) acc = fmaf(h[j], w_hh[gi * HID + j], acc);
        g[gi] = acc;
    }
#pragma unroll
    for (int j = 0; j < HID; ++j) {
        float ig = sigm(g[j]);
        float fg = sigm(g[HID + j]);
        float gg = tanh_fast(g[2 * HID + j]);
        float og = sigm(g[3 * HID + j]);
        float cc = fg * c[j] + ig * gg;
        hcat[b * (2 * HID) + HID + j] = og * tanh_fast(cc);
    }
}

// ---------------------------------------------------------------------------
// Kernel 4: output FC  out[b][d] = hcat[b][:] . fc_w[d][:] + fc_b[d]
// ---------------------------------------------------------------------------
__global__ __launch_bounds__(256) void fc_out(
    const float* __restrict__ hcat,  // [BATCH][10]
    const float* __restrict__ fc_w,  // [DDIM][10]
    const float* __restrict__ fc_b,  // [DDIM]
    float* __restrict__ out)         // [BATCH][DDIM]
{
    const int d = blockIdx.x * blockDim.x + threadIdx.x;
    const int b = blockIdx.y;
    if (d >= DDIM) return;
    const float* hr = hcat + (size_t)b * (2 * HID);
    const float* wr = fc_w + (size_t)d * (2 * HID);
    float acc = fc_b[d];
#pragma unroll
    for (int j = 0; j < 2 * HID; ++j) acc = fmaf(hr[j], wr[j], acc);
    out[(size_t)b * DDIM + d] = acc;
}

// ---------------------------------------------------------------------------
extern "C" void kernel_launch(void* const* d_in, const int* in_sizes, int n_in,
                              void* d_out, int out_size, void* d_ws, size_t ws_size,
                              hipStream_t stream) {
    const float* X      = (const float*)d_in[0];
    const float* h0     = (const float*)d_in[1];
    const float* c0     = (const float*)d_in[2];
    const float* w_ih_f = (const float*)d_in[3];
    const float* w_hh_f = (const float*)d_in[4];
    const float* b_ih_f = (const float*)d_in[5];
    const float* b_hh_f = (const float*)d_in[6];
    const float* w_ih_b = (const float*)d_in[7];
    const float* w_hh_b = (const float*)d_in[8];
    const float* b_ih_b = (const float*)d_in[9];
    const float* b_hh_b = (const float*)d_in[10];
    const float* fc_w   = (const float*)d_in[11];
    const float* fc_b   = (const float*)d_in[12];

    float* ws   = (float*)d_ws;
    float* xp   = ws;                                   // SEQ*BATCH*GATES
    float* gb   = xp + (size_t)SEQ * BATCH * GATES;     // BATCH*GATES
    float* hcat = gb + (size_t)BATCH * GATES;           // BATCH*10

    dim3 blk(256);
    // forward input projection for all timesteps (HBM-bound, WMMA)
    gate_gemm<<<dim3(10, SEQ), blk, 0, stream>>>(X, w_ih_f, b_ih_f, b_hh_f, xp, -1);
    // backward input projection: only t = S-1 needed
    gate_gemm<<<dim3(10, 1), blk, 0, stream>>>(X, w_ih_b, b_ih_b, b_hh_b, gb, SEQ - 1);
    // sequential forward scan, thread per batch row
    lstm_scan_fwd<<<dim3((BATCH + 255) / 256), blk, 0, stream>>>(xp, h0, c0, w_hh_f, hcat);
    // single backward cell step
    bwd_cell<<<dim3((BATCH + 255) / 256), blk, 0, stream>>>(gb, h0, c0, w_hh_b, hcat);
    // output projection
    fc_out<<<dim3((DDIM + 255) / 256, BATCH), blk, 0, stream>>>(hcat, fc_w, fc_b, (float*)d_out);
}